// DetuningEstimation_63651415326813
// MI455X (gfx1250) — compile-verified
//
#include <hip/hip_runtime.h>
#include <hip/hip_bf16.h>

// ---------------------------------------------------------------------------
// Detuning estimation for MI455X (gfx1250, wave32, WMMA).
// FFT-16384 = four-step (128x128) with 128-point DFTs as f32 WMMA matmuls.
// ---------------------------------------------------------------------------

#define NFFT      16384
#define HOPSZ     8192
#define NFREQ     8192
#define TFRAMES   2049
#define FPB       4              // frames per workgroup (register accumulation)
#define AUDIO_LEN 16777216
#define M_UNKNOWN 8188
#define PI_F      3.14159265358979323846f

typedef float v2f __attribute__((ext_vector_type(2)));
typedef float v8f __attribute__((ext_vector_type(8)));

// H = 22050 / 16384 (uniform FFT-bin spacing in Hz)
#define H_BIN (22050.0f / 16384.0f)

// ---------------------------------------------------------------------------
// Kernel 0: precompute frame-invariant tables (all L2-resident, 320 KB total):
//   Fr/Fi : 128-point DFT basis (Re/Im of W_128^{kn}, forward convention)
//   Tr/Ti : four-step twiddles exp(-2*pi*i*k2*n1 / 16384)
//   Wtab  : periodic Hann window
// and zero the log-power accumulator (re-zeroed every call: graph replay).
// ---------------------------------------------------------------------------
__global__ void dft_init_kernel(float* __restrict__ Fr, float* __restrict__ Fi,
                                float* __restrict__ Tr, float* __restrict__ Ti,
                                float* __restrict__ Wtab,
                                float* __restrict__ spec) {
  int idx = blockIdx.x * blockDim.x + threadIdx.x;
  if (idx < NFFT) {
    const int k = idx >> 7, n = idx & 127;
    float sn, cs;
    __sincosf(-2.0f * PI_F * (float)(k * n) / 128.0f, &sn, &cs);
    Fr[idx] = cs;
    Fi[idx] = sn;
    __sincosf(-2.0f * PI_F * (float)(k * n) / (float)NFFT, &sn, &cs);
    Tr[idx] = cs;
    Ti[idx] = sn;
    Wtab[idx] = 0.5f - 0.5f * __cosf(2.0f * PI_F * (float)idx / (float)NFFT);
  }
  if (idx < NFREQ) spec[idx] = 0.0f;
}

// ---------------------------------------------------------------------------
// Kernel 1: one workgroup per FPB frames. Per frame: load + reflect-pad +
// window into LDS as a 128x128 matrix A[n2][n1] (n = n1 + 128*n2), then:
//   Stage 1: B = F128 * A            (2 real WMMA matmuls: Br, Bi)
//   Stage 2: C = B .* Tw[k2*n1]      (elementwise twiddle, in place)
//   Stage 3: D = C * F128[:, 0:64]   (4 real WMMA matmuls) -> X[k2 + 128*k1]
// log(100*|X|^2 + 1) is accumulated in registers across the FPB frames
// (each wave's stage-3 tiles map to fixed bins), then one atomicAdd per bin.
// ---------------------------------------------------------------------------
__global__ __launch_bounds__(256) void spectrogram_kernel(
    const float* __restrict__ audio, const float* __restrict__ Fr,
    const float* __restrict__ Fi, const float* __restrict__ Tr,
    const float* __restrict__ Ti, const float* __restrict__ Wtab,
    float* __restrict__ spec) {
  extern __shared__ float smem[];
  float* A  = smem;              // 16384 floats (64 KB)
  float* Br = smem + 16384;      // 16384 floats
  float* Bi = smem + 32768;      // 16384 floats

  const int tid  = threadIdx.x;
  const int wave = tid >> 5;
  const int lane = tid & 31;
  const int m    = lane & 15;          // row (A-frag) / col (B-frag, C/D col)
  const int koff = (lane >> 4) * 2;    // K offset for upper half-wave
  const int crow = (lane >> 4) * 8;    // C/D row offset for upper half-wave

  v8f sacc[4];                         // per-bin log-power accumulators
#pragma unroll
  for (int t = 0; t < 4; ++t) sacc[t] = (v8f){0, 0, 0, 0, 0, 0, 0, 0};

  for (int fidx = 0; fidx < FPB; ++fidx) {
    const int frame = blockIdx.x * FPB + fidx;
    if (frame >= TFRAMES) break;       // uniform per block: EXEC stays full
    const int base = frame * HOPSZ - (NFFT / 2);   // reflect-pad origin

    // ---- load frame with reflect padding + precomputed Hann window -------
    for (int n = tid; n < NFFT; n += 256) {
      int g = base + n;
      if (g < 0) g = -g;
      if (g >= AUDIO_LEN) g = 2 * AUDIO_LEN - 2 - g;
      A[n] = audio[g] * Wtab[n];   // A[n2*128+n1] == A[n], n = n1 + 128*n2
    }
    __syncthreads();   // also guarantees previous frame's stage 3 finished

    // ---- Stage 1: B = F * A (A real) ------------------------------------
    for (int t = wave; t < 64; t += 8) {
      const int ti = (t >> 3) * 16;   // k2 block
      const int tj = (t & 7) * 16;    // n1 block
      v8f accR = {0.0f, 0.0f, 0.0f, 0.0f, 0.0f, 0.0f, 0.0f, 0.0f};
      v8f accI = accR;
      for (int k = 0; k < 128; k += 4) {
        v2f ar, ai, b;
        const int fb = (ti + m) * 128 + k + koff;   // A-frag: (M=ti+m, K)
        ar.x = Fr[fb];  ar.y = Fr[fb + 1];
        ai.x = Fi[fb];  ai.y = Fi[fb + 1];
        const int bb = (k + koff) * 128 + tj + m;   // B-frag: (K, N=tj+m)
        b.x = A[bb];    b.y = A[bb + 128];
        accR = __builtin_amdgcn_wmma_f32_16x16x4_f32(
            false, ar, false, b, (short)0, accR, false, false);
        accI = __builtin_amdgcn_wmma_f32_16x16x4_f32(
            false, ai, false, b, (short)0, accI, false, false);
      }
#pragma unroll
      for (int r = 0; r < 8; ++r) {
        const int o = (ti + crow + r) * 128 + tj + m;
        Br[o] = accR[r];
        Bi[o] = accI[r];
      }
    }
    __syncthreads();

    // ---- Stage 2: twiddle C = B .* Tw (table lookup), in place -----------
    for (int idx = tid; idx < NFFT; idx += 256) {
      const float cs = Tr[idx], sn = Ti[idx];
      const float br = Br[idx], bi = Bi[idx];
      Br[idx] = br * cs - bi * sn;
      Bi[idx] = br * sn + bi * cs;
    }
    __syncthreads();

    // ---- Stage 3: D = C * F[:, 0:64]; bin = k2 + 128*k1, k1 < 64 ---------
    for (int tl = 0; tl < 4; ++tl) {
      const int t  = wave + 8 * tl;
      const int ti = (t >> 2) * 16;   // k2 block
      const int tj = (t & 3) * 16;    // k1 block (< 64)
      v8f dR = {0.0f, 0.0f, 0.0f, 0.0f, 0.0f, 0.0f, 0.0f, 0.0f};
      v8f dI = dR;
      for (int k = 0; k < 128; k += 4) {
        v2f cr, ci, nci, fr, fi;
        const int cb = (ti + m) * 128 + k + koff;   // A-frag from C
        cr.x = Br[cb];      cr.y = Br[cb + 1];
        ci.x = Bi[cb];      ci.y = Bi[cb + 1];
        nci.x = -ci.x;      nci.y = -ci.y;          // f32 WMMA has no A-neg
        const int fb = (k + koff) * 128 + tj + m;   // B-frag from F
        fr.x = Fr[fb];      fr.y = Fr[fb + 128];
        fi.x = Fi[fb];      fi.y = Fi[fb + 128];
        dR = __builtin_amdgcn_wmma_f32_16x16x4_f32(
            false, cr, false, fr, (short)0, dR, false, false);
        dR = __builtin_amdgcn_wmma_f32_16x16x4_f32(
            false, nci, false, fi, (short)0, dR, false, false);
        dI = __builtin_amdgcn_wmma_f32_16x16x4_f32(
            false, cr, false, fi, (short)0, dI, false, false);
        dI = __builtin_amdgcn_wmma_f32_16x16x4_f32(
            false, ci, false, fr, (short)0, dI, false, false);
      }
#pragma unroll
      for (int r = 0; r < 8; ++r) {
        const float p = dR[r] * dR[r] + dI[r] * dI[r];
        sacc[tl][r] += __logf(fmaf(100.0f, p, 1.0f));
      }
    }
  }

  // ---- single atomic per bin per block (4x fewer atomics than per-frame) --
#pragma unroll
  for (int tl = 0; tl < 4; ++tl) {
    const int t  = wave + 8 * tl;
    const int ti = (t >> 2) * 16;
    const int tj = (t & 3) * 16;
#pragma unroll
    for (int r = 0; r < 8; ++r) {
      const int k2 = ti + crow + r;
      const int k1 = tj + m;
      atomicAdd(&spec[k2 + 128 * k1], sacc[tl][r]);
    }
  }
}

// ---------------------------------------------------------------------------
// Kernel 2: not-a-knot cubic-spline second derivatives M[0:8192] via the
// Thomas algorithm, parallelized as two blocked affine-composition scans
// (the elimination coefficients have the closed form
//   inv_den[i] = (1 - rho^{i+1}) / (s - r*rho^{i+1}),  s=2+sqrt3, r=2-sqrt3,
//   rho = r^2), so the "sequential" recurrences become parallel scans.
// ---------------------------------------------------------------------------
__device__ __forceinline__ float invden_coef(int i) {
  // rho = 7 - 4*sqrt(3); log(rho) = -2.63391579...
  const float t = __expf(-2.6339157938496336f * (float)(i + 1));
  return (1.0f - t) / (3.7320508075688772f - 0.26794919243112270f * t);
}

__global__ __launch_bounds__(256) void spline_kernel(
    const float* __restrict__ y, float* __restrict__ M) {
  extern __shared__ float smem[];
  float* dp = smem;          // 8192
  float* cA = smem + 8192;   // 256
  float* cB = smem + 8448;   // 256
  const int tid = threadIdx.x;
  const float inv_h2 = 1.0f / (H_BIN * H_BIN);

  auto dfun = [&](int j) {
    return (y[j] - 2.0f * y[j + 1] + y[j + 2]) * inv_h2;
  };

  float alpha[32], beta[32];

  // ---- forward sweep: dp_i = (-a_i*inv_i)*dp_{i-1} + rhs_i*inv_i --------
  {
    float Ac = 1.0f, Bc = 0.0f;
#pragma unroll
    for (int e = 0; e < 32; ++e) {
      const int i = tid * 32 + e;
      float a = 1.0f, b = 0.0f;   // identity for out-of-range
      if (i < M_UNKNOWN) {
        float rhs = 6.0f * dfun(i + 1);
        if (i == 0) rhs -= dfun(0);                          // -M1
        if (i == M_UNKNOWN - 1) rhs -= dfun(M_UNKNOWN + 1);  // -Mn2
        const float inv = invden_coef(i);
        a = (i == 0) ? 0.0f : -inv;
        b = rhs * inv;
      }
      alpha[e] = a; beta[e] = b;
      Bc = a * Bc + b;   // compose f_i after accumulated map
      Ac = a * Ac;
    }
    cA[tid] = Ac; cB[tid] = Bc;
  }
  __syncthreads();
  for (int off = 1; off < 256; off <<= 1) {
    const float a1 = cA[tid], b1 = cB[tid];
    float a0 = 1.0f, b0 = 0.0f;
    if (tid >= off) { a0 = cA[tid - off]; b0 = cB[tid - off]; }
    __syncthreads();
    cA[tid] = a1 * a0;
    cB[tid] = a1 * b0 + b1;
    __syncthreads();
  }
  {
    float prev = (tid == 0) ? 0.0f : cB[tid - 1];   // exclusive prefix @ dp=0
#pragma unroll
    for (int e = 0; e < 32; ++e) {
      const int i = tid * 32 + e;
      prev = alpha[e] * prev + beta[e];
      if (i < M_UNKNOWN) dp[i] = prev;
    }
  }
  __syncthreads();

  // ---- backward sweep (reversed order): x_i = -cp_i*x_{i+1} + dp_i ------
  {
    float Ac = 1.0f, Bc = 0.0f;
#pragma unroll
    for (int e = 0; e < 32; ++e) {
      const int j = tid * 32 + e;
      float a = 1.0f, b = 0.0f;
      if (j < M_UNKNOWN) {
        const int i = M_UNKNOWN - 1 - j;
        const float cp = (i < M_UNKNOWN - 1) ? invden_coef(i) : 0.0f;
        a = -cp;
        b = dp[i];
      }
      alpha[e] = a; beta[e] = b;
      Bc = a * Bc + b;
      Ac = a * Ac;
    }
    __syncthreads();
    cA[tid] = Ac; cB[tid] = Bc;
  }
  __syncthreads();
  for (int off = 1; off < 256; off <<= 1) {
    const float a1 = cA[tid], b1 = cB[tid];
    float a0 = 1.0f, b0 = 0.0f;
    if (tid >= off) { a0 = cA[tid - off]; b0 = cB[tid - off]; }
    __syncthreads();
    cA[tid] = a1 * a0;
    cB[tid] = a1 * b0 + b1;
    __syncthreads();
  }
  {
    float xnext = (tid == 0) ? 0.0f : cB[tid - 1];
#pragma unroll
    for (int e = 0; e < 32; ++e) {
      const int j = tid * 32 + e;
      xnext = alpha[e] * xnext + beta[e];
      if (j < M_UNKNOWN) M[2 + (M_UNKNOWN - 1 - j)] = xnext;  // Min -> M[2..]
    }
  }
  __syncthreads();
  if (tid == 0) {
    const float M1  = dfun(0);
    const float Mn2 = dfun(NFREQ - 3);   // d[8189]
    M[1] = M1;
    M[NFREQ - 2] = Mn2;
    M[0] = 2.0f * M1 - M[2];                      // 2*M1  - Min[0]
    M[NFREQ - 1] = 2.0f * Mn2 - M[NFREQ - 3];     // 2*Mn2 - Min[last]
  }
}

// ---------------------------------------------------------------------------
// Kernel 3: spline evaluation on the cent grid, 101-tap mean filter, ReLU,
// dilated all-ones detector conv (dilation 100, pad 50, first 100 outputs),
// argmax -> detuning. One workgroup.
// ---------------------------------------------------------------------------
__global__ __launch_bounds__(512) void detect_kernel(
    const float* __restrict__ y, const float* __restrict__ M,
    const float* __restrict__ w_smooth, const float* __restrict__ w_detect,
    float* __restrict__ out) {
  extern __shared__ float smem[];
  float* cent = smem;           // 8400
  float* filt = smem + 8400;    // 8400
  float* sc   = smem + 16800;   // 128
  const int tid = threadIdx.x;
  const float h = H_BIN;

  for (int j = tid; j < 8400; j += 512) {
    const float note = 24.0f + 0.01f * (float)j;
    const float f = 440.0f * exp2f((note - 69.0f) * (1.0f / 12.0f));
    int i = (int)floorf(f / h);
    i = min(max(i, 0), NFREQ - 2);
    const float t = f - (float)i * h;
    const float u = h - t;
    const float Mi = M[i], Mi1 = M[i + 1], yi = y[i], yi1 = y[i + 1];
    cent[j] = (Mi * u * u * u + Mi1 * t * t * t) / (6.0f * h)
            + (yi - Mi * h * h / 6.0f) * u / h
            + (yi1 - Mi1 * h * h / 6.0f) * t / h;
  }
  __syncthreads();

  for (int j = tid; j < 8400; j += 512) {
    float s = 0.0f;
    for (int q = 0; q < 101; ++q) {
      const int idx = j - 50 + q;
      if (idx >= 0 && idx < 8400) s += cent[idx] * w_smooth[q];
    }
    const float v = cent[j] - s;
    filt[j] = v > 0.0f ? v : 0.0f;
  }
  __syncthreads();

  if (tid < 100) {
    float s = 0.0f;
    for (int q = 0; q < 84; ++q) {
      const int idx = tid - 50 + 100 * q;
      if (idx >= 0 && idx < 8400) s += filt[idx] * w_detect[q];
    }
    sc[tid] = s;
  }
  __syncthreads();

  if (tid == 0) {
    int best = 0;
    float bv = sc[0];
    for (int p = 1; p < 100; ++p)
      if (sc[p] > bv) { bv = sc[p]; best = p; }
    ((int*)out)[0] = best - 50;   // int32 detuning in element 0
  }
  if (tid < 100) out[1 + tid] = sc[tid];
}

// ---------------------------------------------------------------------------
// Launch
// ---------------------------------------------------------------------------
extern "C" void kernel_launch(void* const* d_in, const int* in_sizes, int n_in,
                              void* d_out, int out_size, void* d_ws,
                              size_t ws_size, hipStream_t stream) {
  (void)in_sizes; (void)n_in; (void)out_size; (void)ws_size;
  const float* audio    = (const float*)d_in[0];
  const float* w_smooth = (const float*)d_in[1];
  const float* w_detect = (const float*)d_in[2];

  float* ws   = (float*)d_ws;
  float* Fr   = ws;                 // 16384 floats: DFT-128 basis (Re)
  float* Fi   = ws + 16384;         // 16384 floats: DFT-128 basis (Im)
  float* Tr   = ws + 32768;         // 16384 floats: twiddle (Re)
  float* Ti   = ws + 49152;         // 16384 floats: twiddle (Im)
  float* Wtab = ws + 65536;         // 16384 floats: Hann window
  float* spec = ws + 81920;         // 8192 floats: log-power accumulator
  float* Mbuf = ws + 90112;         // 8192 floats: spline 2nd derivatives

  const size_t lds_spec   = (size_t)3 * 16384 * sizeof(float);   // 192 KB
  const size_t lds_spline = (size_t)(8192 + 512) * sizeof(float);
  const size_t lds_detect = (size_t)(16800 + 128) * sizeof(float);

  // CDNA5 allows up to 320 KB LDS per workgroup; raise the dynamic cap.
  (void)hipFuncSetAttribute((const void*)spectrogram_kernel,
                            hipFuncAttributeMaxDynamicSharedMemorySize,
                            (int)lds_spec);
  (void)hipFuncSetAttribute((const void*)detect_kernel,
                            hipFuncAttributeMaxDynamicSharedMemorySize,
                            (int)lds_detect);

  dft_init_kernel<<<64, 256, 0, stream>>>(Fr, Fi, Tr, Ti, Wtab, spec);
  spectrogram_kernel<<<(TFRAMES + FPB - 1) / FPB, 256, lds_spec, stream>>>(
      audio, Fr, Fi, Tr, Ti, Wtab, spec);
  spline_kernel<<<1, 256, lds_spline, stream>>>(spec, Mbuf);
  detect_kernel<<<1, 512, lds_detect, stream>>>(spec, Mbuf, w_smooth, w_detect,
                                                (float*)d_out);
}